// IF_diff_line_18622978195629
// MI455X (gfx1250) — compile-verified
//
#include <hip/hip_runtime.h>
#include <stdint.h>

// IF-neuron temporal scan (T=8), bandwidth-bound:  ~604 MB traffic @ 23.3 TB/s -> ~26 us floor.
// Load path: CDNA5 async global->LDS DMA (GLOBAL_LOAD_ASYNC_TO_LDS_B128, ASYNCcnt).
// Store path: nontemporal b128 stores so the 302 MB output stream doesn't thrash L2.

typedef float v4f __attribute__((ext_vector_type(4)));

#define T_STEPS 8
#define NSLICE  (T_STEPS + 1)   // 9 time slices including t=0
#define BLOCK   256             // 8 wave32s per block

__global__ __launch_bounds__(BLOCK) void if_scan_kernel(
    const float* __restrict__ x,       // [(T+1)*B, 256,32,32] flattened
    const float* __restrict__ thresh,  // scalar
    float* __restrict__ out,           // same shape as x
    long long n4)                      // float4 elements per time slice
{
    // Per-lane staging slots: each lane async-DMAs its own 16B from each of the
    // 9 slices into its own LDS slot, then reads it back. No barriers needed.
    __shared__ v4f sbuf[NSLICE][BLOCK];   // 9*256*16 = 36864 B

    const int tid = threadIdx.x;
    const long long vbase = (long long)blockIdx.x * BLOCK + tid;
    if (vbase >= n4) return;

    const v4f* xv  = (const v4f*)x;
    v4f*       ov  = (v4f*)out;

    // Issue all 9 async global->LDS loads (tracked by ASYNCcnt, in-order completion).
#pragma unroll
    for (int t = 0; t < NSLICE; ++t) {
        uint64_t gaddr = (uint64_t)(uintptr_t)(xv + (long long)t * n4 + vbase);
        // flat-shared address keeps the LDS byte offset in addr[31:0]
        uint32_t laddr = (uint32_t)(uintptr_t)&sbuf[t][tid];
        asm volatile("global_load_async_to_lds_b128 %0, %1, off"
                     :: "v"(laddr), "v"(gaddr)
                     : "memory");
    }

    const float th = *thresh;  // overlap scalar fetch with the async DMAs

    // Wait for all async transfers to land in LDS.
    asm volatile("s_wait_asynccnt 0" ::: "memory");

    const v4f x0 = sbuf[0][tid];

    const v4f zero = {0.0f, 0.0f, 0.0f, 0.0f};
    // slice 0 of the output is all zeros
    __builtin_nontemporal_store(zero, ov + vbase);

    v4f mem = {0.5f * th, 0.5f * th, 0.5f * th, 0.5f * th};
    v4f ei = zero;   // exp_in accumulator
    v4f eo = zero;   // exp_out accumulator

#pragma unroll
    for (int t = 0; t < T_STEPS; ++t) {
        const v4f xt = sbuf[t + 1][tid];
        const float inv = 1.0f / (float)(t + 1);  // compile-time constant reciprocal
        v4f spike;
#pragma unroll
        for (int c = 0; c < 4; ++c) {
            float m  = mem[c] + xt[c] + ei[c] - eo[c];
            // zif(m - th)*th - zif(-m)*th   (heaviside: u >= 0, incl. -0.0 >= 0)
            float sp = (((m - th) >= 0.0f) ? th : 0.0f)
                     - (((-m)     >= 0.0f) ? th : 0.0f);
            m        = m - sp;
            ei[c]    = ei[c] + (xt[c] - x0[c]) * inv;
            eo[c]    = eo[c] + sp * inv;
            mem[c]   = m;
            spike[c] = sp;
        }
        __builtin_nontemporal_store(spike, ov + (long long)(t + 1) * n4 + vbase);
    }
}

extern "C" void kernel_launch(void* const* d_in, const int* in_sizes, int n_in,
                              void* d_out, int out_size, void* d_ws, size_t ws_size,
                              hipStream_t stream) {
    const float* x      = (const float*)d_in[0];
    const float* thresh = (const float*)d_in[1];
    float*       out    = (float*)d_out;

    const long long total = (long long)in_sizes[0];   // (T+1)*B*256*32*32
    const long long slice = total / NSLICE;           // elements per time slice
    const long long n4    = slice / 4;                // float4 per slice (exact: slice % 4 == 0)
    const long long grid  = (n4 + BLOCK - 1) / BLOCK;

    if_scan_kernel<<<(unsigned)grid, BLOCK, 0, stream>>>(x, thresh, out, n4);
}